// GModule_56581899158045
// MI455X (gfx1250) — compile-verified
//
#include <hip/hip_runtime.h>
#include <math.h>
#include <stdint.h>

#define D 256
#define NR 384
#define LN_EPS 1e-5f
#define SINK_ITERS 20

typedef __attribute__((ext_vector_type(2))) float v2f;
typedef __attribute__((ext_vector_type(8))) float v8f;

// D = A(16x4) * B(4x16) + C   — pure fp32 WMMA on CDNA5
__device__ __forceinline__ v8f wmma4(v2f a, v2f b, v8f c) {
  return __builtin_amdgcn_wmma_f32_16x16x4_f32(false, a, false, b, (short)0, c,
                                               false, false);
}

__device__ __forceinline__ float wave_sum(float v) {
  #pragma unroll
  for (int m = 16; m >= 1; m >>= 1) v += __shfl_xor(v, m, 32);
  return v;
}
__device__ __forceinline__ float wave_max(float v) {
  #pragma unroll
  for (int m = 16; m >= 1; m >>= 1) v = fmaxf(v, __shfl_xor(v, m, 32));
  return v;
}

// ---------------------------------------------------------------------------
// Async DMA copy of a 16xD fp32 tile (16 KB) global -> LDS using
// GLOBAL_LOAD_ASYNC_TO_LDS_B128 (ASYNCcnt-tracked), then wait + barrier.
// Must be called by all 256 threads of the block.
// ---------------------------------------------------------------------------
__device__ __forceinline__ void tile16_async_load(float* sdst,
                                                  const float* __restrict__ gsrc) {
  const int tid = threadIdx.x;
  const uint32_t lbase = (uint32_t)(uintptr_t)sdst;  // LDS byte offset (low 32b of flat)
  #pragma unroll
  for (int it = 0; it < 4; ++it) {
    const int chunk = tid + it * 256;                // 1024 chunks of 16 B
    uint32_t loff = lbase + chunk * 16;
    uint64_t g = (uint64_t)(uintptr_t)gsrc + (uint64_t)chunk * 16;
    asm volatile("global_load_async_to_lds_b128 %0, %1, off"
                 :: "v"(loff), "v"(g) : "memory");
  }
  asm volatile("s_wait_asynccnt 0" ::: "memory");
  __syncthreads();
}

// ---------------------------------------------------------------------------
// Generic C(384x256) = A(384x256) @ W(256x256) + bias   (bias may be null)
// grid = 24 (16-row tiles), block = 256 (8 waves, 2 col-tiles each, shared A)
// ---------------------------------------------------------------------------
__global__ void gemm_kernel(const float* __restrict__ A,
                            const float* __restrict__ W,
                            const float* __restrict__ bias,
                            float* __restrict__ C) {
  __shared__ float sA[16 * D];
  const int row0 = blockIdx.x * 16;
  tile16_async_load(sA, A + row0 * D);
  const int tid = threadIdx.x;
  const int wave = tid >> 5, lane = tid & 31;
  const int half = lane >> 4, l = lane & 15;
  const int col0 = wave * 32;
  v8f acc0 = {}, acc1 = {};
  for (int k = 0; k < D; k += 4) {
    v2f a, b0, b1;
    a.x = sA[l * D + k + 2 * half + 0];
    a.y = sA[l * D + k + 2 * half + 1];
    const float* wr0 = &W[(k + 2 * half + 0) * D];
    const float* wr1 = &W[(k + 2 * half + 1) * D];
    b0.x = wr0[col0 + l];      b0.y = wr1[col0 + l];
    b1.x = wr0[col0 + 16 + l]; b1.y = wr1[col0 + 16 + l];
    acc0 = wmma4(a, b0, acc0);
    acc1 = wmma4(a, b1, acc1);
  }
  const float bv0 = bias ? bias[col0 + l] : 0.f;
  const float bv1 = bias ? bias[col0 + 16 + l] : 0.f;
  #pragma unroll
  for (int i = 0; i < 8; ++i) {
    const int r = row0 + i + 8 * half;
    C[r * D + col0 + l] = acc0[i] + bv0;
    C[r * D + col0 + 16 + l] = acc1[i] + bv1;
  }
}

// ---------------------------------------------------------------------------
// n = LayerNorm(x @ W + b) for both node sets.  grid = 48, block = 256
// ---------------------------------------------------------------------------
__global__ void proj_ln_kernel(const float* __restrict__ n1,
                               const float* __restrict__ n2,
                               const float* __restrict__ W,
                               const float* __restrict__ bias,
                               float* __restrict__ out) {
  __shared__ float sA[16 * D];
  __shared__ float sO[16 * D];
  const int bid = blockIdx.x;
  const int mat = bid / 24, rt = bid % 24;
  const float* A = mat ? n2 : n1;
  float* Og = out + mat * NR * D;
  const int row0 = rt * 16;
  tile16_async_load(sA, A + row0 * D);
  const int tid = threadIdx.x;
  const int wave = tid >> 5, lane = tid & 31;
  const int half = lane >> 4, l = lane & 15;
  const int col0 = wave * 32;
  v8f acc0 = {}, acc1 = {};
  for (int k = 0; k < D; k += 4) {
    v2f a, b0, b1;
    a.x = sA[l * D + k + 2 * half + 0];
    a.y = sA[l * D + k + 2 * half + 1];
    const float* wr0 = &W[(k + 2 * half + 0) * D];
    const float* wr1 = &W[(k + 2 * half + 1) * D];
    b0.x = wr0[col0 + l];      b0.y = wr1[col0 + l];
    b1.x = wr0[col0 + 16 + l]; b1.y = wr1[col0 + 16 + l];
    acc0 = wmma4(a, b0, acc0);
    acc1 = wmma4(a, b1, acc1);
  }
  #pragma unroll
  for (int i = 0; i < 8; ++i) {
    const int r = i + 8 * half;
    sO[r * D + col0 + l] = acc0[i] + bias[col0 + l];
    sO[r * D + col0 + 16 + l] = acc1[i] + bias[col0 + 16 + l];
  }
  __syncthreads();
  // LayerNorm: each wave handles 2 rows
  for (int rr = 0; rr < 2; ++rr) {
    const int r = wave * 2 + rr;
    float s = 0.f, s2 = 0.f;
    for (int c = lane; c < D; c += 32) {
      float x = sO[r * D + c];
      s += x; s2 += x * x;
    }
    s = wave_sum(s); s2 = wave_sum(s2);
    const float mean = s / (float)D;
    const float var = s2 / (float)D - mean * mean;
    const float inv = rsqrtf(var + LN_EPS);
    for (int c = lane; c < D; c += 32)
      Og[(row0 + r) * D + c] = (sO[r * D + c] - mean) * inv;
  }
}

// ---------------------------------------------------------------------------
// Fused attention tail: out = LN( softmax(Q K^T / 16) V Wo + bo + resid )
// grid = 24 (16-row tiles of Q), block = 256
// ---------------------------------------------------------------------------
__global__ void attn_kernel(const float* __restrict__ Q,
                            const float* __restrict__ Kg,
                            const float* __restrict__ V,
                            const float* __restrict__ Wo,
                            const float* __restrict__ bo,
                            const float* __restrict__ resid,
                            float* __restrict__ out) {
  __shared__ float sQ[16 * D];    // Q tile
  __shared__ float sL[16 * NR];   // logits/probs, later U (first 16*D)
  __shared__ float sO[16 * D];    // attn @ V
  const int row0 = blockIdx.x * 16;
  const int tid = threadIdx.x;
  const int wave = tid >> 5, lane = tid & 31;
  const int half = lane >> 4, l = lane & 15;

  tile16_async_load(sQ, Q + row0 * D);

  // logits L = Q K^T * (1/16): 24 col-tiles, 3 per wave, shared A fragment
  {
    const int colb = wave * 48;
    v8f acc0 = {}, acc1 = {}, acc2 = {};
    for (int k = 0; k < D; k += 4) {
      v2f a, b0, b1, b2;
      a.x = sQ[l * D + k + 2 * half + 0];
      a.y = sQ[l * D + k + 2 * half + 1];
      // K^T[k][n] = K[n][k]: per-lane consecutive pair -> b64 load
      b0.x = Kg[(colb + l) * D + k + 2 * half + 0];
      b0.y = Kg[(colb + l) * D + k + 2 * half + 1];
      b1.x = Kg[(colb + 16 + l) * D + k + 2 * half + 0];
      b1.y = Kg[(colb + 16 + l) * D + k + 2 * half + 1];
      b2.x = Kg[(colb + 32 + l) * D + k + 2 * half + 0];
      b2.y = Kg[(colb + 32 + l) * D + k + 2 * half + 1];
      acc0 = wmma4(a, b0, acc0);
      acc1 = wmma4(a, b1, acc1);
      acc2 = wmma4(a, b2, acc2);
    }
    #pragma unroll
    for (int i = 0; i < 8; ++i) {
      const int r = i + 8 * half;
      sL[r * NR + colb + l] = acc0[i] * 0.0625f;
      sL[r * NR + colb + 16 + l] = acc1[i] * 0.0625f;
      sL[r * NR + colb + 32 + l] = acc2[i] * 0.0625f;
    }
  }
  __syncthreads();

  // row softmax over 384 (wave handles 2 rows)
  for (int rr = 0; rr < 2; ++rr) {
    const int r = wave * 2 + rr;
    float m = -INFINITY;
    for (int c = lane; c < NR; c += 32) m = fmaxf(m, sL[r * NR + c]);
    m = wave_max(m);
    float s = 0.f;
    for (int c = lane; c < NR; c += 32) {
      float e = expf(sL[r * NR + c] - m);
      sL[r * NR + c] = e;
      s += e;
    }
    s = wave_sum(s);
    const float inv = 1.f / s;
    for (int c = lane; c < NR; c += 32) sL[r * NR + c] *= inv;
  }
  __syncthreads();

  // O = P @ V  (K-dim = 384), 2 col-tiles per wave, shared A fragment
  {
    const int col0 = wave * 32;
    v8f acc0 = {}, acc1 = {};
    for (int k = 0; k < NR; k += 4) {
      v2f a, b0, b1;
      a.x = sL[l * NR + k + 2 * half + 0];
      a.y = sL[l * NR + k + 2 * half + 1];
      const float* vr0 = &V[(k + 2 * half + 0) * D];
      const float* vr1 = &V[(k + 2 * half + 1) * D];
      b0.x = vr0[col0 + l];      b0.y = vr1[col0 + l];
      b1.x = vr0[col0 + 16 + l]; b1.y = vr1[col0 + 16 + l];
      acc0 = wmma4(a, b0, acc0);
      acc1 = wmma4(a, b1, acc1);
    }
    #pragma unroll
    for (int i = 0; i < 8; ++i) {
      const int r = i + 8 * half;
      sO[r * D + col0 + l] = acc0[i];
      sO[r * D + col0 + 16 + l] = acc1[i];
    }
  }
  __syncthreads();

  // U = O @ Wo + bo + resid  (into sL scratch)
  float* sU = sL;
  {
    const int col0 = wave * 32;
    v8f acc0 = {}, acc1 = {};
    for (int k = 0; k < D; k += 4) {
      v2f a, b0, b1;
      a.x = sO[l * D + k + 2 * half + 0];
      a.y = sO[l * D + k + 2 * half + 1];
      const float* wr0 = &Wo[(k + 2 * half + 0) * D];
      const float* wr1 = &Wo[(k + 2 * half + 1) * D];
      b0.x = wr0[col0 + l];      b0.y = wr1[col0 + l];
      b1.x = wr0[col0 + 16 + l]; b1.y = wr1[col0 + 16 + l];
      acc0 = wmma4(a, b0, acc0);
      acc1 = wmma4(a, b1, acc1);
    }
    #pragma unroll
    for (int i = 0; i < 8; ++i) {
      const int r = i + 8 * half;
      const int c0 = col0 + l, c1 = col0 + 16 + l;
      sU[r * D + c0] = acc0[i] + bo[c0] + resid[(row0 + r) * D + c0];
      sU[r * D + c1] = acc1[i] + bo[c1] + resid[(row0 + r) * D + c1];
    }
  }
  __syncthreads();

  // LayerNorm -> out
  for (int rr = 0; rr < 2; ++rr) {
    const int r = wave * 2 + rr;
    float s = 0.f, s2 = 0.f;
    for (int c = lane; c < D; c += 32) {
      float x = sU[r * D + c];
      s += x; s2 += x * x;
    }
    s = wave_sum(s); s2 = wave_sum(s2);
    const float mean = s / (float)D;
    const float var = s2 / (float)D - mean * mean;
    const float inv = rsqrtf(var + LN_EPS);
    for (int c = lane; c < D; c += 32)
      out[(row0 + r) * D + c] = (sU[r * D + c] - mean) * inv;
  }
}

// ---------------------------------------------------------------------------
// Transpose 384x256 -> 256x384.  grid = (16, 24), block = (16, 16)
// ---------------------------------------------------------------------------
__global__ void transpose_kernel(const float* __restrict__ in,
                                 float* __restrict__ outT) {
  __shared__ float t[16][17];
  const int r = blockIdx.y * 16 + threadIdx.y;
  const int c = blockIdx.x * 16 + threadIdx.x;
  t[threadIdx.y][threadIdx.x] = in[r * D + c];
  __syncthreads();
  const int rT = blockIdx.x * 16 + threadIdx.y;
  const int cT = blockIdx.y * 16 + threadIdx.x;
  outT[rT * NR + cT] = t[threadIdx.x][threadIdx.y];
}

// ---------------------------------------------------------------------------
// M[i][j] = sum_c relu(hx[i][c] + hyT[c][j]) * w2[c] + b2, plus global
// sum / sumsq accumulation for instance norm.  grid = 384, block = 384
// ---------------------------------------------------------------------------
__global__ void affinity_kernel(const float* __restrict__ hx,
                                const float* __restrict__ hyT,
                                const float* __restrict__ w2,
                                const float* __restrict__ b2,
                                float* __restrict__ M,
                                float* __restrict__ stats) {
  __shared__ float sx[D];
  __shared__ float sw[D];
  __shared__ float red1[12];
  __shared__ float red2[12];
  const int i = blockIdx.x;
  const int j = threadIdx.x;  // 384 threads, one output column each
  for (int c = j; c < D; c += 384) {
    sx[c] = hx[i * D + c];
    sw[c] = w2[c];
  }
  __syncthreads();
  float s = b2[0];
  for (int c = 0; c < D; ++c) {
    const float h = sx[c] + hyT[c * NR + j];
    s += fmaxf(h, 0.f) * sw[c];
  }
  M[i * NR + j] = s;
  // block reduction of sum / sumsq -> global atomics
  const int wave = j >> 5, lane = j & 31;
  float ws = wave_sum(s);
  float w2s = wave_sum(s * s);
  if (lane == 0) { red1[wave] = ws; red2[wave] = w2s; }
  __syncthreads();
  if (j == 0) {
    float a = 0.f, b = 0.f;
    for (int w = 0; w < 12; ++w) { a += red1[w]; b += red2[w]; }
    atomicAdd(&stats[0], a);
    atomicAdd(&stats[1], b);
  }
}

__global__ void zero_stats_kernel(float* __restrict__ stats) {
  if (threadIdx.x < 2) stats[threadIdx.x] = 0.f;
}

// ---------------------------------------------------------------------------
// Instance norm + 20 Sinkhorn log iterations + exp.  Single block, 32 waves.
// ---------------------------------------------------------------------------
__global__ void sinkhorn_kernel(float* __restrict__ M,
                                const float* __restrict__ stats,
                                float* __restrict__ P) {
  const int tid = threadIdx.x;
  const int wave = tid >> 5, lane = tid & 31;
  const float n = (float)(NR * NR);
  const float mean = stats[0] / n;
  const float var = stats[1] / n - mean * mean;
  const float inv = rsqrtf(var + LN_EPS);
  for (int idx = tid; idx < NR * NR; idx += 1024)
    M[idx] = (M[idx] - mean) * inv;
  __threadfence();
  __syncthreads();

  for (int it = 0; it < SINK_ITERS; ++it) {
    // rows
    for (int r = wave; r < NR; r += 32) {
      float m = -INFINITY;
      for (int c = lane; c < NR; c += 32) m = fmaxf(m, M[r * NR + c]);
      m = wave_max(m);
      float s = 0.f;
      for (int c = lane; c < NR; c += 32) s += expf(M[r * NR + c] - m);
      s = wave_sum(s);
      const float lse = m + logf(s);
      for (int c = lane; c < NR; c += 32) M[r * NR + c] -= lse;
    }
    __threadfence();
    __syncthreads();
    // cols
    for (int c0 = wave; c0 < NR; c0 += 32) {
      float m = -INFINITY;
      for (int r = lane; r < NR; r += 32) m = fmaxf(m, M[r * NR + c0]);
      m = wave_max(m);
      float s = 0.f;
      for (int r = lane; r < NR; r += 32) s += expf(M[r * NR + c0] - m);
      s = wave_sum(s);
      const float lse = m + logf(s);
      for (int r = lane; r < NR; r += 32) M[r * NR + c0] -= lse;
    }
    __threadfence();
    __syncthreads();
  }
  for (int idx = tid; idx < NR * NR; idx += 1024)
    P[idx] = expf(M[idx]);
}

// ---------------------------------------------------------------------------
extern "C" void kernel_launch(void* const* d_in, const int* in_sizes, int n_in,
                              void* d_out, int out_size, void* d_ws,
                              size_t ws_size, hipStream_t stream) {
  (void)in_sizes; (void)n_in; (void)out_size; (void)ws_size;
  const float* nodes_1 = (const float*)d_in[0];
  const float* nodes_2 = (const float*)d_in[1];
  const float* w_in   = (const float*)d_in[2];
  const float* b_in   = (const float*)d_in[3];
  const float* wq_i = (const float*)d_in[4];  const float* bq_i = (const float*)d_in[5];
  const float* wk_i = (const float*)d_in[6];  const float* bk_i = (const float*)d_in[7];
  const float* wv_i = (const float*)d_in[8];  const float* bv_i = (const float*)d_in[9];
  const float* wo_i = (const float*)d_in[10]; const float* bo_i = (const float*)d_in[11];
  const float* wq_c = (const float*)d_in[12]; const float* bq_c = (const float*)d_in[13];
  const float* wk_c = (const float*)d_in[14]; const float* bk_c = (const float*)d_in[15];
  const float* wv_c = (const float*)d_in[16]; const float* bv_c = (const float*)d_in[17];
  const float* wo_c = (const float*)d_in[18]; const float* bo_c = (const float*)d_in[19];
  const float* w_aff1 = (const float*)d_in[20];
  const float* b_aff1 = (const float*)d_in[21];
  const float* w_aff2 = (const float*)d_in[22];
  const float* b_aff2 = (const float*)d_in[23];

  float* ws = (float*)d_ws;
  const int ND = NR * D;                 // 98304
  float* nA   = ws;                      // n1, n2 after input proj+LN
  float* nB   = ws + 2 * ND;             // n1', n2' after self-MHA
  float* Qb   = ws + 4 * ND;
  float* Kb   = ws + 5 * ND;
  float* Vb   = ws + 6 * ND;
  float* Mb   = ws + 7 * ND;             // 384x384 affinity
  float* hx   = Mb + NR * NR;
  float* hyR  = hx + ND;
  float* hyT  = hyR + ND;
  float* stats = hyT + ND;               // 2 floats

  float* out   = (float*)d_out;
  float* P_out = out;                    // 384*384
  float* n1c   = out + NR * NR;          // 384*256
  float* n2c   = n1c + ND;               // 384*256

  zero_stats_kernel<<<1, 32, 0, stream>>>(stats);

  // input projection + LN for both node sets
  proj_ln_kernel<<<48, 256, 0, stream>>>(nodes_1, nodes_2, w_in, b_in, nA);

  auto mha = [&](const float* qin, const float* kvin,
                 const float* wq, const float* bq, const float* wk,
                 const float* bk, const float* wv, const float* bv,
                 const float* wo, const float* bo, float* o) {
    gemm_kernel<<<24, 256, 0, stream>>>(qin,  wq, bq, Qb);
    gemm_kernel<<<24, 256, 0, stream>>>(kvin, wk, bk, Kb);
    gemm_kernel<<<24, 256, 0, stream>>>(kvin, wv, bv, Vb);
    attn_kernel<<<24, 256, 0, stream>>>(Qb, Kb, Vb, wo, bo, qin, o);
  };

  // self attention (shared weights)
  mha(nA,      nA,      wq_i, bq_i, wk_i, bk_i, wv_i, bv_i, wo_i, bo_i, nB);
  mha(nA + ND, nA + ND, wq_i, bq_i, wk_i, bk_i, wv_i, bv_i, wo_i, bo_i, nB + ND);
  // cross attention -> directly into d_out (n1c, n2c)
  mha(nB,      nB + ND, wq_c, bq_c, wk_c, bk_c, wv_c, bv_c, wo_c, bo_c, n1c);
  mha(nB + ND, nB,      wq_c, bq_c, wk_c, bk_c, wv_c, bv_c, wo_c, bo_c, n2c);

  // affinity projections: hx = n1c @ w1x + b1 ; hy = n2c @ w1y
  gemm_kernel<<<24, 256, 0, stream>>>(n1c, w_aff1,          b_aff1, hx);
  gemm_kernel<<<24, 256, 0, stream>>>(n2c, w_aff1 + D * D,  nullptr, hyR);
  transpose_kernel<<<dim3(16, 24), dim3(16, 16), 0, stream>>>(hyR, hyT);

  affinity_kernel<<<NR, NR, 0, stream>>>(hx, hyT, w_aff2, b_aff2, Mb, stats);
  sinkhorn_kernel<<<1, 1024, 0, stream>>>(Mb, stats, P_out);
}